// ResGCN_43361989821085
// MI455X (gfx1250) — compile-verified
//
#include <hip/hip_runtime.h>

typedef __attribute__((ext_vector_type(2))) float v2f;
typedef __attribute__((ext_vector_type(8))) float v8f;

#define NPTS 12000
#define C 64
#define KNN 16
#define TWO_C 128

// ---------------------------------------------------------------------------
// Kernel 1: batch-masked brute-force kNN within sorted batch segments.
// One thread per point; binary-search own batch segment; sorted top-16 in regs.
// ---------------------------------------------------------------------------
__global__ void knn_kernel(const float* __restrict__ pos,
                           const int* __restrict__ batch,
                           int* __restrict__ knn_idx, int n) {
    int i = blockIdx.x * blockDim.x + threadIdx.x;
    if (i >= n) return;

    int b = batch[i];
    // lower_bound(b)
    int lo = 0, hi = n;
    while (lo < hi) { int m = (lo + hi) >> 1; if (batch[m] < b) lo = m + 1; else hi = m; }
    int s = lo;
    // upper_bound(b)
    lo = 0; hi = n;
    while (lo < hi) { int m = (lo + hi) >> 1; if (batch[m] <= b) lo = m + 1; else hi = m; }
    int e = lo;

    float px = pos[3 * i + 0], py = pos[3 * i + 1], pz = pos[3 * i + 2];
    float si = px * px + py * py + pz * pz;

    float bd[KNN];
    int   bi[KNN];
#pragma unroll
    for (int q = 0; q < KNN; ++q) { bd[q] = __builtin_inff(); bi[q] = i; }

    for (int j = s; j < e; ++j) {
        float qx = pos[3 * j + 0], qy = pos[3 * j + 1], qz = pos[3 * j + 2];
        float sj  = qx * qx + qy * qy + qz * qz;
        float dot = px * qx + py * qy + pz * qz;
        float d   = si + sj - 2.0f * dot;       // same formula as reference
        if (d < bd[KNN - 1]) {
            // stable sorted insert (equal distances keep earlier index first)
            bool placed = false;
#pragma unroll
            for (int q = KNN - 1; q >= 1; --q) {
                bool sh = d < bd[q - 1];
                if (!placed) {
                    bd[q] = sh ? bd[q - 1] : d;
                    bi[q] = sh ? bi[q - 1] : j;
                }
                placed = placed || !sh;
            }
            if (!placed) { bd[0] = d; bi[0] = j; }
        }
    }
#pragma unroll
    for (int q = 0; q < KNN; ++q) knn_idx[i * KNN + q] = bi[q];
}

// ---------------------------------------------------------------------------
// Kernel 2: fold the (2C x C) EdgeConv weight into Wcat (C x 2C) stored in a
// K-pair-interleaved layout so WMMA B-fragments load as a single b64:
//   element (k, col) -> wcat[(k>>1)*2*TWO_C + col*2 + (k&1)]
// where value(k, col) = W[k][col] - W[C+k][col]    (col <  C, P-part)
//                     = W[C+k][col-C]              (col >= C, Q-part)
// ---------------------------------------------------------------------------
__global__ void build_wcat(const float* __restrict__ W, float* __restrict__ wcat) {
    int id = blockIdx.x * blockDim.x + threadIdx.x;   // 0 .. 8191
    if (id >= C * TWO_C) return;
    int k = id >> 7;          // 0..63
    int col = id & 127;       // 0..127
    int d = col & 63;
    float wt = W[k * C + d];
    float wb = W[(C + k) * C + d];
    float v  = (col < C) ? (wt - wb) : wb;
    wcat[(k >> 1) * (2 * TWO_C) + col * 2 + (k & 1)] = v;
}

// ---------------------------------------------------------------------------
// Kernel 3: PQ = x @ Wcat (+bias on first 64 cols), fp32 WMMA 16x16x4.
// Block = 256 threads (8 waves). Block computes a 16 x 128 strip of PQ.
// Wcat (32 KB, pair-interleaved) + x-tile (4 KB) staged in LDS; each wave
// owns one 16-col output tile. A and B fragments are single ds b64 loads.
// ---------------------------------------------------------------------------
__global__ void gemm_pq(const float* __restrict__ x,
                        const float* __restrict__ wcat,
                        const float* __restrict__ bias,
                        float* __restrict__ pq, int n) {
    __shared__ float lw[C * TWO_C];   // pair-interleaved 64 x 128
    __shared__ float lx[16 * C];      // 16 x 64 row-major

    int t = threadIdx.x;
    int m0 = blockIdx.x * 16;

    // stage Wcat: 8192 floats = 2048 float4, 8 per thread (coalesced copy)
    {
        const float4* wsrc = (const float4*)wcat;
        float4* wdst = (float4*)lw;
#pragma unroll
        for (int r = 0; r < 8; ++r) wdst[t + 256 * r] = wsrc[t + 256 * r];
    }
    // stage x tile: 16 rows x 64 = 1024 floats = 256 float4, 1 per thread
    {
        const float4* xsrc = (const float4*)(x + (size_t)m0 * C);
        ((float4*)lx)[t] = xsrc[t];
    }
    __syncthreads();

    int wave = t >> 5;
    int lane = t & 31;
    int half = lane >> 4;       // 0: lanes 0-15, 1: lanes 16-31
    int r    = lane & 15;
    int n0   = wave * 16;       // output column tile

    const v2f* lwp = (const v2f*)lw;   // [kpair][col] -> (W[2kp][col], W[2kp+1][col])

    v8f acc = {0.f, 0.f, 0.f, 0.f, 0.f, 0.f, 0.f, 0.f};

#pragma unroll
    for (int k0 = 0; k0 < C; k0 += 4) {
        int ka = k0 + 2 * half;   // even: lanes 0-15 take (k0,k0+1), 16-31 take (k0+2,k0+3)
        // A 16x4 f32 fragment: contiguous (K=ka, ka+1) pair -> one b64 load
        v2f a = *(const v2f*)&lx[r * C + ka];
        // B 4x16 f32 fragment: pair-interleaved -> one b64 load, no shuffles
        v2f bf = lwp[(ka >> 1) * TWO_C + n0 + r];
        acc = __builtin_amdgcn_wmma_f32_16x16x4_f32(
            false, a, false, bf, (short)0, acc, false, false);
    }

    // C/D 16x16 f32 layout: VGPR v, lanes 0-15 -> row v, lanes 16-31 -> row v+8
    int col = n0 + r;
    float bv = (col < C) ? bias[col] : 0.f;
#pragma unroll
    for (int v = 0; v < 8; ++v) {
        int row = m0 + v + 8 * half;
        pq[(size_t)row * TWO_C + col] = acc[v] + bv;
    }
}

// ---------------------------------------------------------------------------
// Kernel 4: out[i][c] = relu(x[i][c] + max_k relu(P[i][c] + Q[knn[i][k]][c]))
// 64 threads per node -> coalesced 256B gathers of Q rows (L2-resident).
// ---------------------------------------------------------------------------
__global__ void aggregate(const float* __restrict__ xin,
                          const float* __restrict__ pq,
                          const int* __restrict__ knn_idx,
                          float* __restrict__ xout, int n) {
    int t = threadIdx.x;
    int c = t & 63;
    int i = blockIdx.x * 4 + (t >> 6);
    if (i >= n) return;

    float p = pq[(size_t)i * TWO_C + c];
    float m = 0.f;   // all relu'd edge features are >= 0
#pragma unroll
    for (int k = 0; k < KNN; ++k) {
        int j = knn_idx[i * KNN + k];
        float q = pq[(size_t)j * TWO_C + C + c];
        float h = p + q;
        h = h > 0.f ? h : 0.f;
        m = m > h ? m : h;
    }
    float o = xin[(size_t)i * C + c] + m;
    xout[(size_t)i * C + c] = o > 0.f ? o : 0.f;
}

// ---------------------------------------------------------------------------
// Launcher
// inputs: 0 pillar_features (N*C f32), 1 pos (N*3 f32), 2 batch_idx (N i32),
//         3 W0, 4 b0, 5 W1, 6 b1, 7 W2, 8 b2
// ---------------------------------------------------------------------------
extern "C" void kernel_launch(void* const* d_in, const int* in_sizes, int n_in,
                              void* d_out, int out_size, void* d_ws, size_t ws_size,
                              hipStream_t stream) {
    const float* feats = (const float*)d_in[0];
    const float* pos   = (const float*)d_in[1];
    const int*   batch = (const int*)d_in[2];
    const float* Ws[3] = {(const float*)d_in[3], (const float*)d_in[5], (const float*)d_in[7]};
    const float* Bs[3] = {(const float*)d_in[4], (const float*)d_in[6], (const float*)d_in[8]};
    float* out = (float*)d_out;

    char* ws = (char*)d_ws;
    const size_t knn_bytes  = (size_t)NPTS * KNN * sizeof(int);        //   768000
    const size_t wcat_bytes = (size_t)C * TWO_C * sizeof(float);       //    32768
    const size_t pq_bytes   = (size_t)NPTS * TWO_C * sizeof(float);    //  6144000
    const size_t x_bytes    = (size_t)NPTS * C * sizeof(float);        //  3072000

    int*   knn_idx = (int*)ws;
    float* wcat    = (float*)(ws + knn_bytes);
    float* pq      = (float*)(ws + knn_bytes + wcat_bytes);
    float* xA      = (float*)(ws + knn_bytes + wcat_bytes + pq_bytes);
    float* xB      = (float*)(ws + knn_bytes + wcat_bytes + pq_bytes + x_bytes);
    (void)ws_size; (void)n_in; (void)in_sizes; (void)out_size;

    // 1) kNN graph (static; computed once)
    knn_kernel<<<(NPTS + 255) / 256, 256, 0, stream>>>(pos, batch, knn_idx, NPTS);

    // 2) three EdgeConv layers
    const float* layer_in[3]  = {feats, xA, xB};
    float*       layer_out[3] = {xA, xB, out};
    for (int l = 0; l < 3; ++l) {
        build_wcat<<<(C * TWO_C + 255) / 256, 256, 0, stream>>>(Ws[l], wcat);
        gemm_pq<<<NPTS / 16, 256, 0, stream>>>(layer_in[l], wcat, Bs[l], pq, NPTS);
        aggregate<<<NPTS / 4, 256, 0, stream>>>(layer_in[l], pq, knn_idx, layer_out[l], NPTS);
    }
}